// CRF_50955492000590
// MI455X (gfx1250) — compile-verified
//
#include <hip/hip_runtime.h>
#include <stdint.h>

// Viterbi CRF decode for MI455X (gfx1250, wave32).
// Tropical (max,+) semiring -> WMMA mathematically inapplicable; the CDNA5
// feature that applies is the async global->LDS engine (ASYNCcnt) for
// double-buffered emission streaming, plus a register-resident transition
// matrix (loop-invariant across all T steps) and VOPD-friendly VALU chains.

#define KTAGS 64
#define NEGV  -10000.0f
#define FINF  3.0e38f

__device__ __forceinline__ void upd(float s, int p, float& b, int& i) {
  // strict > keeps the FIRST index of the max within this accumulator's stride
  if (s > b) { b = s; i = p; }
}
__device__ __forceinline__ void mrg(float& b, int& i, float b2, int i2) {
  // prefer lower index on exact ties -> global first-index argmax (jnp semantics)
  if (b2 > b || (b2 == b && i2 < i)) { b = b2; i = i2; }
}

__global__ __launch_bounds__(32) void viterbi_crf_kernel(
    const float* __restrict__ feats,    // [B, T, K]
    const float* __restrict__ weights,  // [K, K], weights[next, prev]
    const int*   __restrict__ lens,     // [B]
    float*       __restrict__ score_out,// [B]
    float*       __restrict__ path_out, // [B, T] (tags as float)
    uint16_t*    __restrict__ bptr,     // [B, T, 32] packed backpointer pairs
    int T)
{
  const int b = blockIdx.x;
  const int l = threadIdx.x;            // 0..31, wave32
  const int len = lens[b];

  __shared__ __align__(16) float fvbuf[KTAGS];
  __shared__ __align__(16) float featbuf[2][KTAGS];

  // init forward variables: NEG everywhere except START=0
  fvbuf[l]      = (l == 0) ? 0.0f : NEGV;
  fvbuf[l + 32] = NEGV;

  // Pin the two transition rows this lane owns (next = l and l+32) in VGPRs.
  float wlo[KTAGS], whi[KTAGS];
#pragma unroll
  for (int p = 0; p < KTAGS; p += 4) {
    float4 a = *(const float4*)(weights + (size_t)l * KTAGS + p);
    wlo[p + 0] = a.x; wlo[p + 1] = a.y; wlo[p + 2] = a.z; wlo[p + 3] = a.w;
    float4 c = *(const float4*)(weights + (size_t)(l + 32) * KTAGS + p);
    whi[p + 0] = c.x; whi[p + 1] = c.y; whi[p + 2] = c.z; whi[p + 3] = c.w;
  }

  const unsigned long long featb_u =
      (unsigned long long)(feats + (size_t)b * T * KTAGS);
  uint16_t* bp16 = bptr + (size_t)b * T * 32;

  const unsigned l8 = (unsigned)l * 8u;
  const unsigned ldsA = (unsigned)(uintptr_t)(&featbuf[0][0]) + l8;
  const unsigned ldsB = (unsigned)(uintptr_t)(&featbuf[1][0]) + l8;

  // Async prefetch of feats row t=0 into featbuf[0]:
  // one global_load_async_to_lds_b64 per lane = 256B = one full K row.
  asm volatile("global_load_async_to_lds_b64 %0, %1, %2"
               :: "v"(ldsA), "v"(l8), "s"(featb_u) : "memory");

  for (int t = 0; t < len; ++t) {
    // Prefetch next emission row (clamped so exactly one extra op is always
    // in flight: s_wait_asynccnt<=1 then guarantees THIS row completed).
    {
      int tn = (t + 1 < T) ? (t + 1) : (T - 1);
      unsigned ldsn = ((t + 1) & 1) ? ldsB : ldsA;
      unsigned voff = (unsigned)tn * (KTAGS * 4u) + l8;
      asm volatile("global_load_async_to_lds_b64 %0, %1, %2"
                   :: "v"(ldsn), "v"(voff), "s"(featb_u) : "memory");
    }

    // Tropical mat-vec: 64 prev states, next = l and l+32 per lane.
    // 8 independent (max,argmax) chains -> 16-deep dependences, VOPD-friendly.
    float b0l = -FINF, b1l = -FINF, b2l = -FINF, b3l = -FINF;
    float b0h = -FINF, b1h = -FINF, b2h = -FINF, b3h = -FINF;
    int   i0l = 0, i1l = 1, i2l = 2, i3l = 3;
    int   i0h = 0, i1h = 1, i2h = 2, i3h = 3;
#pragma unroll
    for (int p = 0; p < KTAGS; p += 4) {
      float4 f = *(const float4*)&fvbuf[p];   // ds_load_b128, broadcast read
      upd(f.x + wlo[p + 0], p + 0, b0l, i0l);
      upd(f.y + wlo[p + 1], p + 1, b1l, i1l);
      upd(f.z + wlo[p + 2], p + 2, b2l, i2l);
      upd(f.w + wlo[p + 3], p + 3, b3l, i3l);
      upd(f.x + whi[p + 0], p + 0, b0h, i0h);
      upd(f.y + whi[p + 1], p + 1, b1h, i1h);
      upd(f.z + whi[p + 2], p + 2, b2h, i2h);
      upd(f.w + whi[p + 3], p + 3, b3h, i3h);
    }
    mrg(b0l, i0l, b1l, i1l); mrg(b2l, i2l, b3l, i3l); mrg(b0l, i0l, b2l, i2l);
    mrg(b0h, i0h, b1h, i1h); mrg(b2h, i2h, b3h, i3h); mrg(b0h, i0h, b2h, i2h);

    // Wait for this step's async emission row (next prefetch stays in flight).
    asm volatile("s_wait_asynccnt 0x1" ::: "memory");
    float flo = featbuf[t & 1][l];
    float fhi = featbuf[t & 1][l + 32];

    // fv update in LDS. Single wave: DS ops are in-order; all broadcast reads
    // above precede these writes in program order -> no barrier needed.
    fvbuf[l]      = b0l + flo;
    fvbuf[l + 32] = b0h + fhi;

    // Packed backpointers: one u16 per lane = (next=l | next=l+32 << 8);
    // a single coalesced 64B store per wave per timestep.
    bp16[(size_t)t * 32 + l] = (uint16_t)((unsigned)i0l | ((unsigned)i0h << 8));
  }

  // Terminal: fv + weights[END=1, prev]; wave-wide argmax (first-index ties).
  float tl = fvbuf[l]      + weights[1 * KTAGS + l];
  float th = fvbuf[l + 32] + weights[1 * KTAGS + l + 32];
  float best = tl; int bidx = l;
  if (th > tl) { best = th; bidx = l + 32; }   // tie -> keep lower index l
#pragma unroll
  for (int off = 16; off > 0; off >>= 1) {
    float ob = __shfl_xor(best, off, 32);
    int   oi = __shfl_xor(bidx, off, 32);
    if (ob > best || (ob == best && oi < bidx)) { best = ob; bidx = oi; }
  }
  if (l == 0) score_out[b] = best;

  // Backtrace. Reference has identity backpointers for t >= len -> path = tag
  // there (store-only tail loop). For t < len the wave fetches the whole
  // packed row cooperatively; the row address depends only on t, so the load
  // of row t-1 is issued BEFORE the shuffle that consumes row t (software
  // pipelining hides one L2 latency per step; rows are L2-resident: 192MB L2).
  int tag = bidx;                               // uniform across the wave
  float* po = path_out + (size_t)b * T;
  for (int t = T - 1; t >= len; --t)
    if (l == 0) po[t] = (float)tag;

  int row = (int)bp16[(size_t)(len - 1) * 32 + l];   // len >= 1 guaranteed
  for (int t = len - 1; t >= 0; --t) {
    if (l == 0) po[t] = (float)tag;
    int nrow = 0;
    if (t > 0) nrow = (int)bp16[(size_t)(t - 1) * 32 + l];  // early issue
    int cand = (tag < 32) ? (row & 0xff) : (row >> 8);
    tag = __shfl(cand, tag & 31, 32);           // tag uniform: src lane has it
    row = nrow;
  }
}

extern "C" void kernel_launch(void* const* d_in, const int* in_sizes, int n_in,
                              void* d_out, int out_size, void* d_ws, size_t ws_size,
                              hipStream_t stream) {
  (void)n_in; (void)out_size; (void)ws_size;
  const float* feats   = (const float*)d_in[0];  // [B, T, K] f32
  const float* weights = (const float*)d_in[1];  // [K, K]    f32
  const int*   lens    = (const int*)d_in[2];    // [B]       i32

  const int K  = KTAGS;
  const int B  = in_sizes[2];
  const int T  = (in_sizes[0] / K) / B;

  float*    score = (float*)d_out;               // [B] path scores
  float*    paths = (float*)d_out + B;           // [B, T] tags as f32
  uint16_t* bptrs = (uint16_t*)d_ws;             // needs B*T*64 bytes (33.5 MB)

  viterbi_crf_kernel<<<dim3(B), dim3(32), 0, stream>>>(
      feats, weights, lens, score, paths, bptrs, T);
}